// SparseGraphAttention_71090298683924
// MI455X (gfx1250) — compile-verified
//
#include <hip/hip_runtime.h>
#include <hip/hip_bf16.h>

// Problem constants (match the reference).
#define GAT_N 50000
#define GAT_E 400000
#define GAT_H 8
#define GAT_FIN 128
#define GAT_FOUT 64
#define GAT_ALPHA 0.2f

#define MT 5  // M-tiles (of 16 nodes) per wave; 3125 = 5 * 625 exactly

typedef float v2f __attribute__((ext_vector_type(2)));
typedef float v8f __attribute__((ext_vector_type(8)));

// ---------------------------------------------------------------------------
// Kernel 1: h[head][n][o] = sum_k x[n][k] * W[head][k][o]   via fp32 WMMA.
// Register-blocked: each wave owns 5 M-tiles (80 nodes) x full 64 columns for
// one head, so each B (W) fragment is reused by 5 WMMAs -> 5x less L2 traffic
// on W. Fuses s = h.a_src, d = h.a_dst via 16-lane shuffle reductions out of
// the accumulator registers.
// 5000 waves total; grids are exact so EXEC is all-ones (WMMA requirement).
// ---------------------------------------------------------------------------
__global__ __launch_bounds__(128) void gat_gemm_wmma(
    const float* __restrict__ x, const float* __restrict__ W,
    const float* __restrict__ a, float* __restrict__ hbuf,
    float* __restrict__ sbuf, float* __restrict__ dbuf) {
  const int waves_per_head = (GAT_N / 16) / MT;  // 625
  const int wave = blockIdx.x * (blockDim.x >> 5) + (threadIdx.x >> 5);
  const int head = wave / waves_per_head;        // consecutive waves share head
  const int tile0 = (wave % waves_per_head) * MT;
  if (head >= GAT_H) return;  // never taken (exact grid), wave-uniform

  const int lane = threadIdx.x & 31;
  const int lo = lane & 15;
  const int hi = lane >> 4;  // 0: K+{0,1}, 1: K+{2,3}

  // A tiles: row = node, this lane covers K pair (hi*2, hi*2+1).
  const float* xr[MT];
#pragma unroll
  for (int t = 0; t < MT; ++t)
    xr[t] = x + (size_t)((tile0 + t) * 16 + lo) * GAT_FIN + hi * 2;
  const float* Wh = W + (size_t)head * GAT_FIN * GAT_FOUT;

  v8f acc[MT][4];
#pragma unroll
  for (int t = 0; t < MT; ++t)
#pragma unroll
    for (int n = 0; n < 4; ++n) acc[t][n] = (v8f){};

  for (int k = 0; k < GAT_FIN; k += 4) {
    v2f av[MT];
#pragma unroll
    for (int t = 0; t < MT; ++t) {  // contiguous pair -> global_load_b64
      av[t].x = xr[t][k];
      av[t].y = xr[t][k + 1];
    }
    // B rows (K = k + hi*2, k + hi*2 + 1), columns split in 4 tiles of 16.
    const float* wk0 = Wh + (size_t)(k + hi * 2) * GAT_FOUT + lo;
    const float* wk1 = wk0 + GAT_FOUT;
#pragma unroll
    for (int n = 0; n < 4; ++n) {
      v2f b;
      b.x = wk0[n * 16];
      b.y = wk1[n * 16];
#pragma unroll
      for (int t = 0; t < MT; ++t)
        acc[t][n] = __builtin_amdgcn_wmma_f32_16x16x4_f32(
            false, av[t], false, b, (short)0, acc[t][n], false, false);
    }
  }

  // a_src / a_dst weights for the fused reductions.
  const float* asrc = a + (size_t)head * 2 * GAT_FOUT;
  const float* adst = asrc + GAT_FOUT;
  const float ws0 = asrc[lo], ws1 = asrc[16 + lo], ws2 = asrc[32 + lo], ws3 = asrc[48 + lo];
  const float wd0 = adst[lo], wd1 = adst[16 + lo], wd2 = adst[32 + lo], wd3 = adst[48 + lo];

#pragma unroll
  for (int t = 0; t < MT; ++t) {
    // Store h tile: C/D layout -> vgpr j holds M = j + 8*hi, N = lo (+16*n).
    float* hbase =
        hbuf + ((size_t)head * GAT_N + (size_t)(tile0 + t) * 16) * GAT_FOUT;
#pragma unroll
    for (int j = 0; j < 8; ++j) {
      float* hrow = hbase + (size_t)(j + hi * 8) * GAT_FOUT + lo;
      hrow[0]  = acc[t][0][j];
      hrow[16] = acc[t][1][j];
      hrow[32] = acc[t][2][j];
      hrow[48] = acc[t][3][j];
    }
#pragma unroll
    for (int j = 0; j < 8; ++j) {
      float ps = acc[t][0][j] * ws0 + acc[t][1][j] * ws1 +
                 acc[t][2][j] * ws2 + acc[t][3][j] * ws3;
      float pd = acc[t][0][j] * wd0 + acc[t][1][j] * wd1 +
                 acc[t][2][j] * wd2 + acc[t][3][j] * wd3;
#pragma unroll
      for (int m = 1; m < 16; m <<= 1) {  // stays inside each 16-lane half
        ps += __shfl_xor(ps, m, 32);
        pd += __shfl_xor(pd, m, 32);
      }
      if (lo == 0) {
        const int node = (tile0 + t) * 16 + j + hi * 8;
        sbuf[(size_t)head * GAT_N + node] = ps;
        dbuf[(size_t)head * GAT_N + node] = pd;
      }
    }
  }
}

// ---------------------------------------------------------------------------
// Kernel 2: edge scatter. One thread per (edge, head, o). A full wave shares
// one (edge, head) pair (wave-uniform since 32 | tid of wave start and 32 | 64), so the
// s/d loads broadcast and the agg atomics are 32 consecutive floats.
// agg is accumulated directly in d_out with layout [n][h][o]. h and agg both
// fit in the 192 MB L2, so gathers/atomics run at L2 rates.
// ---------------------------------------------------------------------------
__global__ __launch_bounds__(256) void gat_edge(
    const int* __restrict__ ei, const float* __restrict__ sbuf,
    const float* __restrict__ dbuf, const float* __restrict__ hbuf,
    float* __restrict__ rowsum, float* __restrict__ agg) {
  const size_t tid = (size_t)blockIdx.x * blockDim.x + threadIdx.x;
  const size_t total = (size_t)GAT_E * GAT_H * GAT_FOUT;
  if (tid >= total) return;
  const int o = (int)(tid & 63);
  const size_t pid = tid >> 6;
  const int head = (int)(pid & (GAT_H - 1));
  const int edge = (int)(pid >> 3);

  const int src = ei[edge];
  const int dst = ei[GAT_E + edge];

  const float logit = sbuf[(size_t)head * GAT_N + src] +
                      dbuf[(size_t)head * GAT_N + dst];
  const float lr = logit > 0.0f ? logit : GAT_ALPHA * logit;
  const float val = __expf(-lr);

  if (o == 0) atomicAdd(rowsum + (size_t)head * GAT_N + src, val);

  const float hv = hbuf[((size_t)head * GAT_N + dst) * GAT_FOUT + o];
  atomicAdd(agg + ((size_t)src * GAT_H + head) * GAT_FOUT + o, val * hv);
}

// ---------------------------------------------------------------------------
// Kernel 3: out[n][h*64+o] /= rowsum[h][n]  (in place on d_out).
// ---------------------------------------------------------------------------
__global__ __launch_bounds__(256) void gat_norm(
    float* __restrict__ out, const float* __restrict__ rowsum) {
  const size_t i = (size_t)blockIdx.x * blockDim.x + threadIdx.x;
  const size_t total = (size_t)GAT_N * GAT_H * GAT_FOUT;
  if (i >= total) return;
  const int n = (int)(i >> 9);           // / 512
  const int head = (int)((i >> 6) & 7);  // / 64 % 8
  out[i] = out[i] / rowsum[(size_t)head * GAT_N + n];
}

extern "C" void kernel_launch(void* const* d_in, const int* in_sizes, int n_in,
                              void* d_out, int out_size, void* d_ws, size_t ws_size,
                              hipStream_t stream) {
  (void)in_sizes; (void)n_in; (void)out_size; (void)ws_size;
  const float* x = (const float*)d_in[0];
  const int* ei = (const int*)d_in[1];
  const float* W = (const float*)d_in[2];
  const float* a = (const float*)d_in[3];
  float* out = (float*)d_out;

  // Workspace layout (bytes): h (102.4MB) | s (1.6MB) | d (1.6MB) | rowsum (1.6MB)
  char* ws = (char*)d_ws;
  float* hbuf = (float*)ws;  // H*N*FOUT
  float* sbuf = (float*)(ws + (size_t)GAT_H * GAT_N * GAT_FOUT * 4);
  float* dbuf = sbuf + (size_t)GAT_H * GAT_N;
  float* rowsum = dbuf + (size_t)GAT_H * GAT_N;

  // Zero accumulators every call (harness poisons buffers; must be deterministic).
  hipMemsetAsync(out, 0, (size_t)GAT_N * GAT_H * GAT_FOUT * sizeof(float), stream);
  hipMemsetAsync(rowsum, 0, (size_t)GAT_H * GAT_N * sizeof(float), stream);

  // 1) WMMA GEMM + fused s/d. 5000 waves (8 heads x 625), 4 waves/block.
  const int nwaves = GAT_H * ((GAT_N / 16) / MT);  // 5000
  gat_gemm_wmma<<<nwaves / 4, 128, 0, stream>>>(x, W, a, hbuf, sbuf, dbuf);

  // 2) Edge scatter: E*H*64 threads.
  const size_t etotal = (size_t)GAT_E * GAT_H * GAT_FOUT;
  gat_edge<<<(unsigned)((etotal + 255) / 256), 256, 0, stream>>>(
      ei, sbuf, dbuf, hbuf, rowsum, out);

  // 3) Normalize in place.
  const size_t ototal = (size_t)GAT_N * GAT_H * GAT_FOUT;
  gat_norm<<<(unsigned)((ototal + 255) / 256), 256, 0, stream>>>(out, rowsum);
}